// encoder_65438121721861
// MI455X (gfx1250) — compile-verified
//
#include <hip/hip_runtime.h>

// GCN forward on MI455X (gfx1250, wave32).
// d_in order: x, edge_index(int64), W1, b1, bn1_g, bn1_b, W, bias, bn2_g, bn2_b
// Output: float32 [N,128].

typedef __attribute__((ext_vector_type(2))) float v2f;
typedef __attribute__((ext_vector_type(8))) float v8f;

#define HDIM 128
#define GCN_EPS 1e-5f
#define GCN_SLOPE 0.01f
#define GCN_ITERS 5

// ---------------------------------------------------------------------------
// Degree / normalization
// ---------------------------------------------------------------------------
__global__ __launch_bounds__(256) void deg_init_kernel(float* deg, int N) {
  int i = blockIdx.x * 256 + threadIdx.x;
  if (i < N) deg[i] = 1.0f;  // self-loop contributes 1 to every node
}

__global__ __launch_bounds__(256) void deg_edges_kernel(const long long* __restrict__ ei,
                                                        float* __restrict__ deg, int E) {
  int e = blockIdx.x * 256 + threadIdx.x;
  if (e < E) atomicAdd(&deg[(int)ei[(size_t)E + e]], 1.0f);
}

__global__ __launch_bounds__(256) void deg_finalize_kernel(float* dinv, int N) {
  int i = blockIdx.x * 256 + threadIdx.x;
  if (i < N) dinv[i] = rsqrtf(dinv[i]);  // deg >= 1 always (self-loops)
}

// ---------------------------------------------------------------------------
// WMMA GEMM: out[N,128] = in[N,128] @ W[128,128], full f32 via 16x16x4 WMMA.
// Block = 256 threads = 8 waves; each wave computes a 16x128 output strip.
// W is staged in LDS *pair-swizzled*: Wp[k/2][col] = {W[k][col], W[k+1][col]}
// so every B fragment is one aligned ds_load_b64 straight into an even VGPR
// pair (no repacking movs between LDS and v_wmma).
// ---------------------------------------------------------------------------
__global__ __launch_bounds__(256) void gemm_wmma_kernel(const float* __restrict__ in,
                                                        const float* __restrict__ W,
                                                        float* __restrict__ out, int N) {
  __shared__ v2f Wp[(HDIM / 2) * HDIM];  // 64KB
  float* Wpf = (float*)Wp;
  const int tid = threadIdx.x;
  // Cooperative swizzle-fill: global reads coalesced, Wp[(k>>1)*128 + c] gets
  // component (k&1) from W[k*128 + c].
  for (int idx = tid; idx < HDIM * HDIM; idx += 256) {
    const int k = idx >> 7;
    const int c = idx & (HDIM - 1);
    Wpf[((k >> 1) * HDIM + c) * 2 + (k & 1)] = W[idx];
  }
  __syncthreads();

  const int wave = tid >> 5;
  const int lane = tid & 31;
  const int half = lane >> 4;   // 0: K+{0,1}, rows 0..7 ; 1: K+{2,3}, rows 8..15
  const int l16  = lane & 15;
  const int rowBase = blockIdx.x * 128 + wave * 16;

  v8f zero = {};
  v8f acc[8];
#pragma unroll
  for (int t = 0; t < 8; ++t) acc[t] = zero;

  // Clamp A-row so loads stay in-bounds without EXEC divergence (WMMA needs EXEC all-1s).
  const int aRow = min(rowBase + l16, N - 1);
  const float* __restrict__ aPtr = in + (size_t)aRow * HDIM + half * 2;

#pragma unroll 4
  for (int k0 = 0; k0 < HDIM; k0 += 4) {
    // A fragment (16x4 f32): lanes 0-15 hold K=k0,k0+1; lanes 16-31 hold K=k0+2,k0+3.
    const v2f a = *(const v2f*)(aPtr + k0);
    // B fragments: pair index kp = (k0 + half*2)/2 -> one b64 load per tile.
    const v2f* __restrict__ bRow = Wp + ((k0 >> 1) + half) * HDIM;
#pragma unroll
    for (int t = 0; t < 8; ++t) {
      const v2f b = bRow[t * 16 + l16];
      acc[t] = __builtin_amdgcn_wmma_f32_16x16x4_f32(
          /*neg_a=*/false, a, /*neg_b=*/false, b,
          /*c_mod=*/(short)0, acc[t], /*reuse_a=*/false, /*reuse_b=*/false);
    }
  }

  // C/D layout: VGPR i -> row i (lanes 0-15) or row 8+i (lanes 16-31), col = lane&15.
#pragma unroll
  for (int t = 0; t < 8; ++t) {
    const int col = t * 16 + l16;
#pragma unroll
    for (int i = 0; i < 8; ++i) {
      const int row = rowBase + half * 8 + i;
      if (row < N) out[(size_t)row * HDIM + col] = acc[t][i];
    }
  }
}

// ---------------------------------------------------------------------------
// Edge scatter: one lane per (edge, 4 columns). agg[dst] += m[src] * dinv[s]*dinv[d].
// Working set (m + agg + edges + dinv ~ 167MB) fits the 192MB L2, so these
// atomics run at L2 rate, not HBM.
// ---------------------------------------------------------------------------
__global__ __launch_bounds__(256) void scatter_edges_kernel(const float* __restrict__ m,
                                                            const long long* __restrict__ ei,
                                                            const float* __restrict__ dinv,
                                                            float* __restrict__ agg, int E) {
  int t = blockIdx.x * 256 + threadIdx.x;
  int e = t >> 5;
  if (e >= E) return;
  const int c4 = (t & 31) * 4;
  const int s = (int)ei[e];
  const int d = (int)ei[(size_t)E + e];
  const float w = dinv[s] * dinv[d];
  const float4 v = *(const float4*)(m + (size_t)s * HDIM + c4);
  float* p = agg + (size_t)d * HDIM + c4;
  atomicAdd(p + 0, v.x * w);
  atomicAdd(p + 1, v.y * w);
  atomicAdd(p + 2, v.z * w);
  atomicAdd(p + 3, v.w * w);
}

// Self loops: each (node, col-quad) touched once; runs after scatter (stream order).
__global__ __launch_bounds__(256) void selfloop_kernel(const float* __restrict__ m,
                                                       const float* __restrict__ dinv,
                                                       float* __restrict__ agg, int N) {
  int t = blockIdx.x * 256 + threadIdx.x;
  int i = t >> 5;
  if (i >= N) return;
  const int c4 = (t & 31) * 4;
  const float w = dinv[i] * dinv[i];
  const size_t off = (size_t)i * HDIM + c4;
  const float4 v = *(const float4*)(m + off);
  float4 a = *(const float4*)(agg + off);
  a.x += v.x * w; a.y += v.y * w; a.z += v.z * w; a.w += v.w * w;
  *(float4*)(agg + off) = a;
}

// ---------------------------------------------------------------------------
// BatchNorm: per-column sums, then fused normalize+affine+LeakyReLU.
// (Pre-BN per-column biases are absorbed by the mean subtraction -> omitted.)
// ---------------------------------------------------------------------------
__global__ __launch_bounds__(128) void bn_stats_kernel(const float* __restrict__ z,
                                                       float* __restrict__ colsum,
                                                       float* __restrict__ colsumsq, int N) {
  const int col = threadIdx.x;
  const int rowsPer = (N + gridDim.x - 1) / gridDim.x;
  const int r0 = blockIdx.x * rowsPer;
  const int r1 = min(N, r0 + rowsPer);
  float s = 0.0f, q = 0.0f;
  for (int r = r0; r < r1; ++r) {
    const float v = z[(size_t)r * HDIM + col];
    s += v;
    q += v * v;
  }
  atomicAdd(&colsum[col], s);
  atomicAdd(&colsumsq[col], q);
}

__global__ __launch_bounds__(256) void bn_apply_kernel(const float* __restrict__ z,
                                                       const float* __restrict__ colsum,
                                                       const float* __restrict__ colsumsq,
                                                       const float* __restrict__ g,
                                                       const float* __restrict__ b,
                                                       float* __restrict__ h, int N) {
  int t = blockIdx.x * 256 + threadIdx.x;
  if (t >= N * 32) return;
  const int c4 = (t & 31) * 4;
  const size_t off = (size_t)(t >> 5) * HDIM + c4;
  const float invN = 1.0f / (float)N;
  const float4 v = *(const float4*)(z + off);
  float4 o;
  float vv[4] = {v.x, v.y, v.z, v.w};
  float oo[4];
#pragma unroll
  for (int j = 0; j < 4; ++j) {
    const int col = c4 + j;
    const float mean = colsum[col] * invN;
    const float var = colsumsq[col] * invN - mean * mean;
    float y = (vv[j] - mean) * rsqrtf(var + GCN_EPS) * g[col] + b[col];
    oo[j] = y > 0.0f ? y : GCN_SLOPE * y;
  }
  o.x = oo[0]; o.y = oo[1]; o.z = oo[2]; o.w = oo[3];
  *(float4*)(h + off) = o;
}

// ---------------------------------------------------------------------------
// Final Frobenius norm + scale into d_out.
// ---------------------------------------------------------------------------
__global__ __launch_bounds__(256) void frob_kernel(const float* __restrict__ h,
                                                   float* __restrict__ out, int n4) {
  float s = 0.0f;
  for (int i = blockIdx.x * 256 + threadIdx.x; i < n4; i += gridDim.x * 256) {
    const float4 v = ((const float4*)h)[i];
    s += v.x * v.x + v.y * v.y + v.z * v.z + v.w * v.w;
  }
  __shared__ float red[256];
  red[threadIdx.x] = s;
  __syncthreads();
  for (int o = 128; o > 0; o >>= 1) {
    if (threadIdx.x < o) red[threadIdx.x] += red[threadIdx.x + o];
    __syncthreads();
  }
  if (threadIdx.x == 0) atomicAdd(out, red[0]);
}

__global__ __launch_bounds__(256) void scale_out_kernel(const float* __restrict__ h,
                                                        const float* __restrict__ sumsq,
                                                        float* __restrict__ out, int n4) {
  int i = blockIdx.x * 256 + threadIdx.x;
  if (i >= n4) return;
  const float inv = 1.0f / sqrtf(*sumsq);
  const float4 v = ((const float4*)h)[i];
  float4 o;
  o.x = v.x * inv; o.y = v.y * inv; o.z = v.z * inv; o.w = v.w * inv;
  ((float4*)out)[i] = o;
}

// ---------------------------------------------------------------------------
extern "C" void kernel_launch(void* const* d_in, const int* in_sizes, int n_in,
                              void* d_out, int out_size, void* d_ws, size_t ws_size,
                              hipStream_t stream) {
  const float* x        = (const float*)d_in[0];
  const long long* ei   = (const long long*)d_in[1];  // int64 edge_index [2,E]
  const float* W1       = (const float*)d_in[2];
  // d_in[3] (b1) and d_in[7] (bias): per-column pre-BN biases, exactly absorbed
  // by BatchNorm's mean subtraction -> intentionally unused.
  const float* bn1_g    = (const float*)d_in[4];
  const float* bn1_b    = (const float*)d_in[5];
  const float* W        = (const float*)d_in[6];
  const float* bn2_g    = (const float*)d_in[8];
  const float* bn2_b    = (const float*)d_in[9];

  const int N = in_sizes[0] / HDIM;
  const int E = in_sizes[1] / 2;

  char* ws = (char*)d_ws;
  size_t off = 0;
  float* h    = (float*)(ws + off); off += (size_t)N * HDIM * sizeof(float);
  float* m    = (float*)(ws + off); off += (size_t)N * HDIM * sizeof(float);
  float* agg  = (float*)(ws + off); off += (size_t)N * HDIM * sizeof(float);
  float* dinv = (float*)(ws + off); off += (((size_t)N * sizeof(float)) + 255) & ~(size_t)255;
  float* colsum   = (float*)(ws + off); off += HDIM * sizeof(float);
  float* colsumsq = (float*)(ws + off); off += HDIM * sizeof(float);
  float* sumsq    = (float*)(ws + off); off += sizeof(float);

  const int n32 = N * 32;  // (node, col-quad) work items
  const int bN   = (N + 255) / 256;
  const int bE   = (E + 255) / 256;
  const int b32  = (n32 + 255) / 256;
  const int bE32 = (int)(((size_t)E * 32 + 255) / 256);
  const int gemmBlocks = (N + 127) / 128;

  // --- symmetric normalization coefficients ---
  deg_init_kernel<<<bN, 256, 0, stream>>>(dinv, N);
  deg_edges_kernel<<<bE, 256, 0, stream>>>(ei, dinv, E);
  deg_finalize_kernel<<<bN, 256, 0, stream>>>(dinv, N);

  // --- input layer: h = leaky(BN(x @ W1)); Frobenius scaling provably cancels ---
  gemm_wmma_kernel<<<gemmBlocks, 256, 0, stream>>>(x, W1, m, N);
  hipMemsetAsync(colsum, 0, 2 * HDIM * sizeof(float), stream);
  bn_stats_kernel<<<512, 128, 0, stream>>>(m, colsum, colsumsq, N);
  bn_apply_kernel<<<b32, 256, 0, stream>>>(m, colsum, colsumsq, bn1_g, bn1_b, h, N);

  // --- propagation iterations ---
  for (int it = 0; it < GCN_ITERS; ++it) {
    gemm_wmma_kernel<<<gemmBlocks, 256, 0, stream>>>(h, W, m, N);
    hipMemsetAsync(agg, 0, (size_t)N * HDIM * sizeof(float), stream);
    scatter_edges_kernel<<<bE32, 256, 0, stream>>>(m, ei, dinv, agg, E);
    selfloop_kernel<<<b32, 256, 0, stream>>>(m, dinv, agg, N);
    hipMemsetAsync(colsum, 0, 2 * HDIM * sizeof(float), stream);
    bn_stats_kernel<<<512, 128, 0, stream>>>(agg, colsum, colsumsq, N);
    bn_apply_kernel<<<b32, 256, 0, stream>>>(agg, colsum, colsumsq, bn2_g, bn2_b, h, N);
  }

  // --- final normalization ---
  hipMemsetAsync(sumsq, 0, sizeof(float), stream);
  frob_kernel<<<2048, 256, 0, stream>>>(h, sumsq, n32);
  scale_out_kernel<<<b32, 256, 0, stream>>>(h, sumsq, (float*)d_out, n32);
}